// SatLossEvaluator_6468220748481
// MI455X (gfx1250) — compile-verified
//
#include <hip/hip_runtime.h>
#include <hip/hip_bf16.h>
#include <stdint.h>

typedef __attribute__((ext_vector_type(2))) float v2f;
typedef __attribute__((ext_vector_type(8))) float v8f;

// Wave32 sum via V_WMMA_F32_16X16X4_F32.
// A (16x4 f32): lanes 0-15 hold {K=0,K=1} for M=lane; lanes 16-31 hold {K=2,K=3}.
// s in vgpr0, 0 in vgpr1, B = ones  =>  D[m][n] = s_m + s_{m+16} (replicated over n).
// Summing the 8 C/D VGPRs gives half the wave per lane-half; xor-16 completes it.
__device__ __forceinline__ float wave_sum_wmma(float s) {
  v2f a; a[0] = s;    a[1] = 0.0f;
  v2f b; b[0] = 1.0f; b[1] = 1.0f;
  v8f c = {};
  c = __builtin_amdgcn_wmma_f32_16x16x4_f32(false, a, false, b, (short)0, c, false, false);
  float r = ((c[0] + c[1]) + (c[2] + c[3])) + ((c[4] + c[5]) + (c[6] + c[7]));
  r += __shfl_xor(r, 16, 32);
  return r;
}

// Block (256 threads = 8 waves) sum; result valid in thread 0.
__device__ __forceinline__ float block_sum(float s, float* wsum) {
  float w = wave_sum_wmma(s);
  const int lane = threadIdx.x & 31;
  const int wid  = threadIdx.x >> 5;
  if (lane == 0) wsum[wid] = w;
  __syncthreads();
  float t = 0.0f;
  if (threadIdx.x == 0) {
    for (int i = 0; i < 8; ++i) t += wsum[i];
  }
  return t;
}

// Stage one 256-clause tile (ef: 3072B, vidx: 3072B) into LDS at ldsbase.
// Every thread issues exactly one b128 and one b64 => ASYNCcnt += 2 per wave,
// EXEC all-ones at both issue sites (required for exact partial waits).
// LDS layout: [0,3072) = ef, [3072,6144) = vidx.
__device__ __forceinline__ void stage_tile_async(const float* ef, const int* vidx,
                                                 int f0, uint32_t ldsbase, int tid) {
  const uint64_t gef = (uint64_t)(uintptr_t)(ef   + (size_t)3 * (size_t)f0);
  const uint64_t gix = (uint64_t)(uintptr_t)(vidx + (size_t)3 * (size_t)f0);
  // b128 chunks: t<192 -> ef bytes [16t,16t+16); t>=192 -> vidx bytes [16(t-192),..)
  const uint64_t g0 = (tid < 192) ? (gef + (uint32_t)(tid * 16))
                                  : (gix + (uint32_t)((tid - 192) * 16));
  const uint32_t l0 = ldsbase + (uint32_t)(tid * 16);
  asm volatile("global_load_async_to_lds_b128 %0, %1, off" :: "v"(l0), "v"(g0) : "memory");
  // b64 chunks: vidx bytes [1024 + 8t, ...), LDS [4096 + 8t, ...)
  const uint64_t g1 = gix + 1024u + (uint32_t)(tid * 8);
  const uint32_t l1 = ldsbase + 4096u + (uint32_t)(tid * 8);
  asm volatile("global_load_async_to_lds_b64 %0, %1, off" :: "v"(l1), "v"(g1) : "memory");
}

__device__ __forceinline__ float clause_term(float e0, float e1, float e2,
                                             float v0, float v1, float v2,
                                             float coeff, float eps, int sharp) {
  const float ev0 = fmaf(e0, v0, (1.0f - e0) * 0.5f);
  const float ev1 = fmaf(e1, v1, (1.0f - e1) * 0.5f);
  const float ev2 = fmaf(e2, v2, (1.0f - e2) * 0.5f);
  const float w0 = __expf(coeff * ev0);
  const float w1 = __expf(coeff * ev1);
  const float w2 = __expf(coeff * ev2);
  const float nom = fmaf(w0, ev0, fmaf(w1, ev1, w2 * ev2));
  const float den = w0 + w1 + w2;
  const float cv = den / fmaxf(nom, eps);
  const float d = cv - 1.0f;
  float p;
  if (sharp == 5) {            // fast path for the reference's loss_sharpness
    const float d2 = d * d;
    p = d2 * d2 * d;
  } else {
    p = 1.0f;
    for (int i = 0; i < sharp; ++i) p *= d;
  }
  return __logf(fmaxf(1.0f + p, eps));
}

__global__ __launch_bounds__(256) void clause_loss_kernel(
    const float* __restrict__ vp,      // variable_prediction [V]
    const int*   __restrict__ vidx,    // graph_map row 0 [E]
    const float* __restrict__ ef,      // edge_feature [E]
    const float* __restrict__ gs,      // global_step [1]
    const float* __restrict__ epsp,    // eps [1]
    const float* __restrict__ maxcp,   // max_coeff [1]
    const int*   __restrict__ sharpp,  // loss_sharpness [1]
    float* __restrict__ partials, int F) {
  __shared__ float4 smraw[2][384];     // double-buffered 6144B tiles
  __shared__ float  wsum[8];

  const int tid = threadIdx.x;
  const int G   = gridDim.x;
  const float eps   = epsp[0];
  const float coeff = fminf(__powf(gs[0], 0.4f), maxcp[0]);
  const int   sharp = sharpp[0];
  const int ntFull  = F >> 8;          // number of fully-populated 256-clause tiles

  const uint32_t ldsA = (uint32_t)(uintptr_t)&smraw[0][0];
  const uint32_t ldsB = (uint32_t)(uintptr_t)&smraw[1][0];

  float acc = 0.0f;
  int t   = blockIdx.x;
  int buf = 0;

  if (t < ntFull) stage_tile_async(ef, vidx, t << 8, ldsA, tid);

  while (t < ntFull) {                 // uniform per block
    const int  tn = t + G;
    const bool hn = (tn < ntFull);     // uniform per block
    __syncthreads();                   // all reads of the other buffer are done
    if (hn) {
      stage_tile_async(ef, vidx, tn << 8, buf ? ldsA : ldsB, tid);
      // 2 newest async ops belong to tile tn; in-order completion => tile t done.
      asm volatile("s_wait_asynccnt 0x2" ::: "memory");
    } else {
      asm volatile("s_wait_asynccnt 0x0" ::: "memory");
    }
    __syncthreads();                   // every wave's wait passed => tile t in LDS

    const float* sm_ef = (const float*)&smraw[buf][0];
    const int*   sm_ix = (const int*)(&smraw[buf][0] + 192);
    const float e0 = sm_ef[3 * tid];
    const float e1 = sm_ef[3 * tid + 1];
    const float e2 = sm_ef[3 * tid + 2];
    const int   i0 = sm_ix[3 * tid];
    const int   i1 = sm_ix[3 * tid + 1];
    const int   i2 = sm_ix[3 * tid + 2];

    acc += clause_term(e0, e1, e2, vp[i0], vp[i1], vp[i2], coeff, eps, sharp);

    buf ^= 1;
    t = tn;
  }

  // Single partial tail tile (if F % 256 != 0), handled by exactly one block.
  if ((ntFull << 8) < F && (int)blockIdx.x == (ntFull % G)) {
    const int f = (ntFull << 8) + tid;
    if (f < F) {
      const size_t e = (size_t)3 * (size_t)f;
      acc += clause_term(ef[e], ef[e + 1], ef[e + 2],
                         vp[vidx[e]], vp[vidx[e + 1]], vp[vidx[e + 2]],
                         coeff, eps, sharp);
    }
  }

  const float tt = block_sum(acc, wsum);
  if (tid == 0) partials[blockIdx.x] = tt;
}

__global__ __launch_bounds__(256) void reduce_finalize_kernel(
    const float* __restrict__ partials, int n, float invF,
    float* __restrict__ out) {
  __shared__ float wsum[8];
  float s = 0.0f;
  for (int i = threadIdx.x; i < n; i += 256) s += partials[i];
  const float t = block_sum(s, wsum);
  if (threadIdx.x == 0) out[0] = t * invF;
}

extern "C" void kernel_launch(void* const* d_in, const int* in_sizes, int n_in,
                              void* d_out, int out_size, void* d_ws, size_t ws_size,
                              hipStream_t stream) {
  const float* vp    = (const float*)d_in[0];
  const int*   gmap  = (const int*)d_in[2];   // row 0 = var_idx, row 1 = fun_idx
  const float* ef    = (const float*)d_in[5];
  const float* gs    = (const float*)d_in[7];
  const float* eps   = (const float*)d_in[8];
  const float* maxc  = (const float*)d_in[9];
  const int*   sharp = (const int*)d_in[10];

  const int F = in_sizes[4];          // batch_function_map length == #clauses
  const int* vidx = gmap;             // first E entries of graph_map

  int nblocks = (F + 255) / 256;
  if (nblocks > 2048) nblocks = 2048;  // partials fit in 8KB of d_ws
  float* partials = (float*)d_ws;

  clause_loss_kernel<<<nblocks, 256, 0, stream>>>(
      vp, vidx, ef, gs, eps, maxc, sharp, partials, F);
  reduce_finalize_kernel<<<1, 256, 0, stream>>>(
      partials, nblocks, 1.0f / (float)F, (float*)d_out);
}